// SparseProjector_30614526886240
// MI455X (gfx1250) — compile-verified
//
#include <hip/hip_runtime.h>
#include <hip/hip_bf16.h>

// Problem constants (match the reference).
#define SRC_SIZE 200000
#define DST_SIZE 50000
#define NUM_EDGES 800000
#define FEAT 32
#define BATCH 4

#define SCAN_T 1024
#define AGG_WAVES 8   // waves (dst nodes) per block in k_aggregate

// CDNA5 async global->LDS path (gfx1250): probe-guarded.
#if defined(__has_builtin)
#if __has_builtin(__builtin_amdgcn_global_load_async_to_lds_b64) && \
    __has_builtin(__builtin_amdgcn_s_wait_asynccnt)
#define USE_ASYNC_LDS 1
#endif
#endif

// Decisive probe: device pass must have the async builtins (compile fails
// loudly if not, instead of silently taking the fallback).
#if defined(__HIP_DEVICE_COMPILE__) && !defined(USE_ASYNC_LDS)
#error "gfx1250 async global->LDS builtins not available on device pass"
#endif

#if defined(USE_ASYNC_LDS)
typedef int v2i_t __attribute__((vector_size(2 * sizeof(int))));
typedef __attribute__((address_space(1))) v2i_t* gbl_v2i_p;
typedef __attribute__((address_space(3))) v2i_t* lds_v2i_p;
#endif

// ---------------------------------------------------------------------------
// K0: zero norm[] (float) and counts[] (int) — contiguous in workspace.
// ---------------------------------------------------------------------------
__global__ void k_zero(int* __restrict__ p, int n) {
    int i = blockIdx.x * blockDim.x + threadIdx.x;
    if (i < n) p[i] = 0;
}

// ---------------------------------------------------------------------------
// K1: per-dst weight sum (norm) and edge count, via L2 atomics.
// ---------------------------------------------------------------------------
__global__ void k_count_norm(const long long* __restrict__ ei,
                             const float* __restrict__ w,
                             float* __restrict__ norm,
                             int* __restrict__ counts) {
    int e = blockIdx.x * blockDim.x + threadIdx.x;
    if (e >= NUM_EDGES) return;
    int d = (int)ei[NUM_EDGES + e];
    atomicAdd(&norm[d], w[e]);
    atomicAdd(&counts[d], 1);
}

// ---------------------------------------------------------------------------
// K2: exclusive prefix sum over counts -> offsets, also seed cursor.
// Single workgroup (1024 threads = 32 waves) Hillis-Steele with carry.
// ---------------------------------------------------------------------------
__global__ void k_scan(const int* __restrict__ counts,
                       int* __restrict__ offsets,
                       int* __restrict__ cursor) {
    __shared__ int sh[SCAN_T];
    __shared__ int carry_sh;
    const int tid = threadIdx.x;
    if (tid == 0) carry_sh = 0;
    __syncthreads();
    for (int base = 0; base < DST_SIZE; base += SCAN_T) {
        int i = base + tid;
        int v = (i < DST_SIZE) ? counts[i] : 0;
        sh[tid] = v;
        __syncthreads();
        #pragma unroll
        for (int off = 1; off < SCAN_T; off <<= 1) {
            int t = (tid >= off) ? sh[tid - off] : 0;
            __syncthreads();
            sh[tid] += t;
            __syncthreads();
        }
        int excl = sh[tid] - v + carry_sh;
        if (i < DST_SIZE) { offsets[i] = excl; cursor[i] = excl; }
        __syncthreads();
        if (tid == 0) carry_sh += sh[SCAN_T - 1];
        __syncthreads();
    }
}

// ---------------------------------------------------------------------------
// K3: bin edges by destination into a packed (src, normalized-w) array.
// ---------------------------------------------------------------------------
__global__ void k_bin(const long long* __restrict__ ei,
                      const float* __restrict__ w,
                      const float* __restrict__ norm,
                      int* __restrict__ cursor,
                      int2* __restrict__ sedge) {
    int e = blockIdx.x * blockDim.x + threadIdx.x;
    if (e >= NUM_EDGES) return;
    int s = (int)ei[e];
    int d = (int)ei[NUM_EDGES + e];
    float wn = w[e] / (norm[d] + 1e-8f);
    int pos = atomicAdd(&cursor[d], 1);
    sedge[pos] = make_int2(s, __float_as_int(wn));
}

// ---------------------------------------------------------------------------
// K4: gather-aggregate. One wave32 per destination node; lane == feature.
// Edge chunks of 32 staged per-wave via global_load_async_to_lds_b64
// (ASYNCcnt), then consumed as uniform ds_load broadcasts; (src, w) pushed
// to SGPRs with readfirstlane so batch gathers are SGPR-base + lane-offset
// loads and v_fmac takes w from an SGPR. Host-pass fallback uses __shfl.
// ---------------------------------------------------------------------------
__global__ void __launch_bounds__(AGG_WAVES * 32)
k_aggregate(const int2* __restrict__ sedge,
            const int* __restrict__ offsets,
            const int* __restrict__ counts,
            const float* __restrict__ x,
            float* __restrict__ out) {
#if defined(USE_ASYNC_LDS) && defined(__HIP_DEVICE_COMPILE__)
    __shared__ int2 ebuf[AGG_WAVES][32];
#endif
    const int lane = threadIdx.x & 31;
    const int wv   = threadIdx.x >> 5;
    const int d    = blockIdx.x * (blockDim.x >> 5) + wv;
    if (d >= DST_SIZE) return;

    const int start = offsets[d];
    const int cnt   = counts[d];

    const size_t bstride = (size_t)SRC_SIZE * FEAT;  // batch stride in x
    const float* __restrict__ x0 = x;
    const float* __restrict__ x1 = x + bstride;
    const float* __restrict__ x2 = x + 2 * bstride;
    const float* __restrict__ x3 = x + 3 * bstride;

    float acc0 = 0.f, acc1 = 0.f, acc2 = 0.f, acc3 = 0.f;

    for (int c = 0; c < cnt; c += 32) {
        int rem = cnt - c;
        if (rem > 32) rem = 32;

        // Prefetch next edge chunk into cache (global_prefetch_b8).
        if (c + 32 < cnt)
            __builtin_prefetch((const void*)&sedge[start + c + 32 + lane], 0, 0);

#if defined(USE_ASYNC_LDS) && defined(__HIP_DEVICE_COMPILE__)
        // Stage this chunk's edge records into per-wave LDS via the CDNA5
        // async copy engine (one b64 per active lane, tracked by ASYNCcnt).
        if (lane < rem) {
            const void* g = (const void*)(sedge + (size_t)start + c + lane);
            void* l = (void*)&ebuf[wv][lane];
            __builtin_amdgcn_global_load_async_to_lds_b64(
                (gbl_v2i_p)(unsigned long long)(uintptr_t)g,
                (lds_v2i_p)(unsigned)(uintptr_t)l,
                0, 0);
        }
        __builtin_amdgcn_s_wait_asynccnt(0);
        __builtin_amdgcn_wave_barrier();

        for (int j = 0; j < rem; ++j) {
            int2 e2 = ebuf[wv][j];                       // uniform LDS broadcast
            int   s  = __builtin_amdgcn_readfirstlane(e2.x);
            float wj = __uint_as_float(
                (unsigned)__builtin_amdgcn_readfirstlane(e2.y));
            const size_t ro = (size_t)s * FEAT + lane;
            acc0 = fmaf(wj, x0[ro], acc0);
            acc1 = fmaf(wj, x1[ro], acc1);
            acc2 = fmaf(wj, x2[ro], acc2);
            acc3 = fmaf(wj, x3[ro], acc3);
        }
#else
        int2 ed = make_int2(0, 0);
        if (lane < rem) ed = sedge[(size_t)start + c + lane];
        for (int j = 0; j < rem; ++j) {
            int   s  = __shfl(ed.x, j, 32);
            float wj = __int_as_float(__shfl(ed.y, j, 32));
            const size_t ro = (size_t)s * FEAT + lane;
            acc0 = fmaf(wj, x0[ro], acc0);
            acc1 = fmaf(wj, x1[ro], acc1);
            acc2 = fmaf(wj, x2[ro], acc2);
            acc3 = fmaf(wj, x3[ro], acc3);
        }
#endif
    }

    const size_t ostride = (size_t)DST_SIZE * FEAT;
    float* op = out + (size_t)d * FEAT + lane;
    op[0 * ostride] = acc0;
    op[1 * ostride] = acc1;
    op[2 * ostride] = acc2;
    op[3 * ostride] = acc3;
}

// ---------------------------------------------------------------------------
// Launch: K0 zero -> K1 count/norm -> K2 scan -> K3 bin -> K4 aggregate.
// Workspace layout (7.2 MB total):
//   norm   : DST floats
//   counts : DST ints
//   offsets: DST ints
//   cursor : DST ints
//   sedge  : E   int2  (8B-aligned: 4*DST*4 = 800000 bytes precede it)
// ---------------------------------------------------------------------------
extern "C" void kernel_launch(void* const* d_in, const int* in_sizes, int n_in,
                              void* d_out, int out_size, void* d_ws, size_t ws_size,
                              hipStream_t stream) {
    (void)in_sizes; (void)n_in; (void)out_size; (void)ws_size;

    const float*     x  = (const float*)d_in[0];
    const long long* ei = (const long long*)d_in[1];
    const float*     w  = (const float*)d_in[2];
    float* out = (float*)d_out;

    char* ws = (char*)d_ws;
    float* norm   = (float*)ws;  ws += (size_t)DST_SIZE * sizeof(float);
    int*   counts = (int*)ws;    ws += (size_t)DST_SIZE * sizeof(int);
    int*   offs   = (int*)ws;    ws += (size_t)DST_SIZE * sizeof(int);
    int*   cursor = (int*)ws;    ws += (size_t)DST_SIZE * sizeof(int);
    int2*  sedge  = (int2*)ws;

    {
        int n = 2 * DST_SIZE;
        k_zero<<<dim3((n + 1023) / 1024), dim3(1024), 0, stream>>>((int*)norm, n);
    }
    k_count_norm<<<dim3((NUM_EDGES + 255) / 256), dim3(256), 0, stream>>>(
        ei, w, norm, counts);
    k_scan<<<dim3(1), dim3(SCAN_T), 0, stream>>>(counts, offs, cursor);
    k_bin<<<dim3((NUM_EDGES + 255) / 256), dim3(256), 0, stream>>>(
        ei, w, norm, cursor, sedge);
    k_aggregate<<<dim3((DST_SIZE + AGG_WAVES - 1) / AGG_WAVES),
                  dim3(AGG_WAVES * 32), 0, stream>>>(
        sedge, offs, counts, x, out);
}